// ChamferDist_24790551233433
// MI455X (gfx1250) — compile-verified
//
#include <hip/hip_runtime.h>

typedef __attribute__((ext_vector_type(2))) float v2f;
typedef __attribute__((ext_vector_type(8))) float v8f;

#define K_PTS 4096
#define COL_TILES (K_PTS / 16)   // 256
#define BATCH 8

__global__ void ChamferDist_zero_kernel(float* out) {
    if (threadIdx.x == 0) out[0] = 0.0f;
}

__global__ __launch_bounds__(256) void ChamferDist_24790551233433_kernel(
    const float* __restrict__ adv,   // [B, K, 3]
    const float* __restrict__ ori,   // [B, K, 3]
    const float* __restrict__ w,     // [B]
    float* __restrict__ out)         // [1]
{
    const int lane      = threadIdx.x & 31;
    const int waveInBlk = threadIdx.x >> 5;
    const int gw        = blockIdx.x * 8 + waveInBlk;   // 0..2047
    const int b         = gw >> 8;                       // batch
    const int rowTile   = gw & 255;                      // 16-row tile of adv
    const int hi        = lane >> 4;                     // half-wave select
    const int ln        = lane & 15;

    const size_t base = (size_t)b * K_PTS * 3;

    // ---- A operand (16x4 f32): row m = rowTile*16 + ln, 4th dim = 1.
    // Layout: lanes 0-15 hold K=0 (VGPR0) / K=1 (VGPR1); lanes 16-31 hold K=2 / K=3.
    const int m = rowTile * 16 + ln;
    const float ax = adv[base + (size_t)m * 3 + 0];
    const float ay = adv[base + (size_t)m * 3 + 1];
    const float az = adv[base + (size_t)m * 3 + 2];
    v2f A;
    A.x = hi ? az   : ax;
    A.y = hi ? 1.0f : ay;

    // ---- C operand (16x16 f32 accum layout): C[r] = |a_{rowTile*16 + 8*hi + r}|^2.
    v8f C;
#pragma unroll
    for (int r = 0; r < 8; ++r) {
        const int row = rowTile * 16 + hi * 8 + r;
        const float x = adv[base + (size_t)row * 3 + 0];
        const float y = adv[base + (size_t)row * 3 + 1];
        const float z = adv[base + (size_t)row * 3 + 2];
        C[r] = x * x + y * y + z * z;
    }

    // running row-min (per held D element)
    v8f mn;
#pragma unroll
    for (int r = 0; r < 8; ++r) mn[r] = 3.4e38f;

    // ---- sweep all 256 ori column tiles; each tile is one V_WMMA_F32_16X16X4_F32
    for (int jt = 0; jt < COL_TILES; ++jt) {
        const int col = jt * 16 + ln;
        const float bx = ori[base + (size_t)col * 3 + 0];
        const float by = ori[base + (size_t)col * 3 + 1];
        const float bz = ori[base + (size_t)col * 3 + 2];
        const float b2 = bx * bx + by * by + bz * bz;
        // B operand (4x16 f32): lane = column N; lanes 0-15 hold K=0/1, lanes 16-31 hold K=2/3.
        v2f Bv;
        Bv.x = hi ? (-2.0f * bz) : (-2.0f * bx);
        Bv.y = hi ? b2           : (-2.0f * by);

        // D = A*B + C  ==  |a|^2 + |b|^2 - 2 a.b  (exact f32, matches reference math)
        v8f D = __builtin_amdgcn_wmma_f32_16x16x4_f32(
            /*neg_a=*/false, A, /*neg_b=*/false, Bv,
            /*c_mod=*/(short)0, C, /*reuse_a=*/false, /*reuse_b=*/false);

#pragma unroll
        for (int r = 0; r < 8; ++r) mn[r] = fminf(mn[r], D[r]);
    }

    // ---- min across the 16 lanes (columns) of each half-wave
#pragma unroll
    for (int msk = 1; msk <= 8; msk <<= 1) {
#pragma unroll
        for (int r = 0; r < 8; ++r)
            mn[r] = fminf(mn[r], __shfl_xor(mn[r], msk, 32));
    }

    // sum the 8 row-mins this half-wave owns (identical in all 16 lanes of the half)
    float s = 0.0f;
#pragma unroll
    for (int r = 0; r < 8; ++r) s += mn[r];
    // combine the two halves (rows 0-7 + rows 8-15)
    s += __shfl_xor(s, 16, 32);

    if (lane == 0) {
        const float scale = w[b] / ((float)K_PTS * (float)BATCH);
        atomicAdd(out, s * scale);
    }
}

extern "C" void kernel_launch(void* const* d_in, const int* in_sizes, int n_in,
                              void* d_out, int out_size, void* d_ws, size_t ws_size,
                              hipStream_t stream) {
    (void)in_sizes; (void)n_in; (void)out_size; (void)d_ws; (void)ws_size;
    const float* adv = (const float*)d_in[0];
    const float* ori = (const float*)d_in[1];
    const float* w   = (const float*)d_in[2];
    float* out = (float*)d_out;

    ChamferDist_zero_kernel<<<1, 32, 0, stream>>>(out);
    // 2048 waves: 8 batches x 256 row tiles; 8 waves (256 threads) per block.
    ChamferDist_24790551233433_kernel<<<256, 256, 0, stream>>>(adv, ori, w, out);
}